// GraphSAGEModel_10479720202336
// MI455X (gfx1250) — compile-verified
//
#include <hip/hip_runtime.h>

#define C_DIM 128
#define BN_EPS 1e-5f

typedef float v2f __attribute__((ext_vector_type(2)));
typedef float v8f __attribute__((ext_vector_type(8)));

__device__ __forceinline__ v8f wmma_f32(v2f a, v2f b, v8f c) {
  // 8 args: (neg_a, A, neg_b, B, c_mod, C, reuse_a, reuse_b)
  return __builtin_amdgcn_wmma_f32_16x16x4_f32(false, a, false, b, (short)0, c,
                                               false, false);
}

// ---------------------------------------------------------------------------
// Edge aggregation: one wave per edge. Lanes cooperatively gather the 128-float
// source row (float4 per lane, fully coalesced; h rows are L2-resident) and
// scatter with f32 atomics (resolved at L2). Lane 0 bumps the degree counter.
// ---------------------------------------------------------------------------
__global__ __launch_bounds__(256) void aggregate_kernel(
    const float* __restrict__ h, const int* __restrict__ src,
    const int* __restrict__ dst, float* __restrict__ agg,
    float* __restrict__ cnt, int E) {
  long tid = (long)blockIdx.x * blockDim.x + threadIdx.x;
  int e = (int)(tid >> 5);
  int l = (int)(tid & 31);
  if (e >= E) return;
  int s = src[e];
  int d = dst[e];
  const float4 v = *(const float4*)(h + (size_t)s * C_DIM + l * 4);
  float* ap = agg + (size_t)d * C_DIM + l * 4;
  atomicAdd(ap + 0, v.x);
  atomicAdd(ap + 1, v.y);
  atomicAdd(ap + 2, v.z);
  atomicAdd(ap + 3, v.w);
  if (l == 0) atomicAdd(cnt + d, 1.0f);
}

// ---------------------------------------------------------------------------
// Fused GEMM + bias + (ReLU/BN) epilogue using V_WMMA_F32_16X16X4_F32.
//
// Weights are staged in LDS ROW-PAIR-INTERLEAVED: sW[pair][col][2] with
// pair = row/2 and pair stride PS = 2*COUT+32 floats. A lane's B fragment
// (rows k+koff, k+koff+1 at one column) is then a single aligned 8-byte word:
// one ds_load_b64 straight into the even-aligned VGPR pair WMMA consumes --
// no assembly movs, immediate offsets (t*128 B), and PS mod 64 = 32 puts the
// two half-waves on disjoint bank groups (conflict-free).
//
// Inner loop per K=4 slice: prefetch next A fragments (global), clause-load
// NT (x2 if DUAL) B fragments via ds_load_b64, then a back-to-back WMMA chain.
// DUAL accumulates agg@Wl + h@Wr into one tile set; mean folded into A load.
// MODE 0: relu->bn (SAGE layers). MODE 1: bn->relu (MLP head).
// ---------------------------------------------------------------------------
template <int COUT, int MODE, bool DUAL>
__global__ __launch_bounds__(256) void gemm_fused(
    const float* __restrict__ A1, const float* __restrict__ cnt,
    const float* __restrict__ W1, const float* __restrict__ A2,
    const float* __restrict__ W2, const float* __restrict__ bias,
    const float* __restrict__ bn_g, const float* __restrict__ bn_b,
    const float* __restrict__ bn_m, const float* __restrict__ bn_v,
    float* __restrict__ out, int rows) {
  constexpr int NT = COUT / 16;
  constexpr int PS = 2 * COUT + 32;  // pair stride in floats (bank shift 32)
  constexpr int NPAIR = C_DIM / 2;
  extern __shared__ float lds[];
  float* sW1 = lds;
  float* sW2 = lds + NPAIR * PS;  // used only when DUAL

  // Stage weights row-pair-interleaved: two coalesced b32 loads -> ds_store_b64.
  for (int p = threadIdx.x; p < NPAIR * COUT; p += 256) {
    const int pair = p / COUT;
    const int col = p % COUT;
    v2f w;
    w.x = W1[(size_t)(2 * pair) * COUT + col];
    w.y = W1[(size_t)(2 * pair + 1) * COUT + col];
    *(v2f*)(sW1 + pair * PS + col * 2) = w;
    if (DUAL) {
      v2f w2;
      w2.x = W2[(size_t)(2 * pair) * COUT + col];
      w2.y = W2[(size_t)(2 * pair + 1) * COUT + col];
      *(v2f*)(sW2 + pair * PS + col * 2) = w2;
    }
  }
  __syncthreads();

  const int lane = threadIdx.x & 31;
  const int wid = threadIdx.x >> 5;
  const int rowBase = blockIdx.x * 128 + wid * 16;
  if (rowBase >= rows) return;  // wave-uniform: EXEC stays all-ones for WMMA

  const int mrow = lane & 15;        // A-frag row / B-frag col within tile
  const int half = lane >> 4;        // which K half-pair this lane covers
  const int koff = half * 2;
  const int arow = rowBase + mrow;

  float sA = 1.0f;
  if (DUAL) sA = 1.0f / fmaxf(cnt[arow], 1.0f);  // fold mean into A load

  const float* a1p = A1 + (size_t)arow * C_DIM + koff;
  const float* a2p = DUAL ? (A2 + (size_t)arow * C_DIM + koff) : nullptr;
  // Lane-fixed LDS bases; per-slice advance = 2 pairs = 2*PS floats.
  const float* lw1 = sW1 + half * PS + mrow * 2;
  const float* lw2 = DUAL ? (sW2 + half * PS + mrow * 2) : nullptr;

  v8f zero = {};
  v8f acc[NT];
#pragma unroll
  for (int t = 0; t < NT; ++t) acc[t] = zero;

  // Software pipeline: A fragments for slice k are loaded one iteration ahead.
  v2f a1 = *(const v2f*)(a1p);
  v2f a2 = {};
  if (DUAL) a2 = *(const v2f*)(a2p);

#pragma unroll 4
  for (int k = 0; k < C_DIM; k += 4) {
    v2f ca1 = a1, ca2 = a2;
    if (k + 4 < C_DIM) {
      a1 = *(const v2f*)(a1p + k + 4);
      if (DUAL) a2 = *(const v2f*)(a2p + k + 4);
    }
    ca1.x *= sA;
    ca1.y *= sA;

    // Clause-load all B fragments for this slice: one ds_load_b64 each.
    v2f b1[NT];
    v2f b2[NT];
#pragma unroll
    for (int t = 0; t < NT; ++t) {
      b1[t] = *(const v2f*)(lw1 + t * 32);
      if (DUAL) b2[t] = *(const v2f*)(lw2 + t * 32);
    }
    lw1 += 2 * PS;
    if (DUAL) lw2 += 2 * PS;

    // WMMA chain, staggered dscnt waits handled by the scheduler.
#pragma unroll
    for (int t = 0; t < NT; ++t) {
      acc[t] = wmma_f32(ca1, b1[t], acc[t]);
      if (DUAL) acc[t] = wmma_f32(ca2, b2[t], acc[t]);
    }
  }

  // Epilogue: C/D layout -> lane holds rows (r + 8*(lane>>4)), col = lane&15.
  const int rOff = half * 8;
#pragma unroll
  for (int t = 0; t < NT; ++t) {
    const int col = t * 16 + mrow;
    const float bi = bias[col];
    const float scale = bn_g[col] * rsqrtf(bn_v[col] + BN_EPS);
    const float mu = bn_m[col];
    const float beta = bn_b[col];
#pragma unroll
    for (int r = 0; r < 8; ++r) {
      const int row = rowBase + rOff + r;
      float val = acc[t][r] + bi;
      if (MODE == 0) {
        val = fmaxf(val, 0.0f);
        val = (val - mu) * scale + beta;
      } else {
        val = (val - mu) * scale + beta;
        val = fmaxf(val, 0.0f);
      }
      if (row < rows) out[(size_t)row * COUT + col] = val;
    }
  }
}

// Final 64 -> 1 projection: one thread per row, float4 row loads.
__global__ __launch_bounds__(256) void head3_kernel(
    const float* __restrict__ h2, const float* __restrict__ w,
    const float* __restrict__ b, float* __restrict__ out, int rows) {
  int i = blockIdx.x * blockDim.x + threadIdx.x;
  if (i >= rows) return;
  const float* hp = h2 + (size_t)i * 64;
  float s = 0.0f;
#pragma unroll
  for (int k = 0; k < 64; k += 4) {
    float4 hv = *(const float4*)(hp + k);
    s += hv.x * w[k + 0] + hv.y * w[k + 1] + hv.z * w[k + 2] + hv.w * w[k + 3];
  }
  out[i] = s + b[0];
}

extern "C" void kernel_launch(void* const* d_in, const int* in_sizes, int n_in,
                              void* d_out, int out_size, void* d_ws,
                              size_t ws_size, hipStream_t stream) {
  const float* x = (const float*)d_in[0];
  const int* edge = (const int*)d_in[1];
  const float* Wl = (const float*)d_in[3];
  const float* bl = (const float*)d_in[4];
  const float* Wr = (const float*)d_in[5];
  const float* bng = (const float*)d_in[6];
  const float* bnb = (const float*)d_in[7];
  const float* bnm = (const float*)d_in[8];
  const float* bnv = (const float*)d_in[9];
  const float* cW1 = (const float*)d_in[10];
  const float* cb1 = (const float*)d_in[11];
  const float* cg1 = (const float*)d_in[12];
  const float* cbe1 = (const float*)d_in[13];
  const float* cm1 = (const float*)d_in[14];
  const float* cv1 = (const float*)d_in[15];
  const float* cW2 = (const float*)d_in[16];
  const float* cb2 = (const float*)d_in[17];
  const float* cg2 = (const float*)d_in[18];
  const float* cbe2 = (const float*)d_in[19];
  const float* cm2 = (const float*)d_in[20];
  const float* cv2 = (const float*)d_in[21];
  const float* cW3 = (const float*)d_in[22];
  const float* cb3 = (const float*)d_in[23];
  float* out = (float*)d_out;

  const int N = in_sizes[0] / C_DIM;  // 100000
  const int E = in_sizes[1] / 2;      // 1600000
  const int BATCH = 50000;            // reference batch_size constant

  // Workspace layout (all L2-friendly fp32):
  float* hA = (float*)d_ws;              // N*128
  float* hB = hA + (size_t)N * C_DIM;    // N*128
  float* agg = hB + (size_t)N * C_DIM;   // N*128
  float* cnt = agg + (size_t)N * C_DIM;  // N
  float* h1 = agg;                       // reuse agg region after layers
  float* h2 = agg + (size_t)BATCH * C_DIM;

  const int gemmBlocks = (N + 127) / 128;
  const long aggThreads = (long)E * 32;
  const int aggBlocks = (int)((aggThreads + 255) / 256);

  // Dynamic LDS: pair-interleaved weight copies (pair stride 2*COUT+32).
  const size_t ldsLayer = 2 * (size_t)(C_DIM / 2) * (2 * C_DIM + 32) * sizeof(float);
  const size_t ldsHead1 = (size_t)(C_DIM / 2) * (2 * C_DIM + 32) * sizeof(float);
  const size_t ldsHead2 = (size_t)(C_DIM / 2) * (2 * 64 + 32) * sizeof(float);

  const float* hin = x;
  float* houts[3] = {hA, hB, hA};
  for (int l = 0; l < 3; ++l) {
    hipMemsetAsync(agg, 0, (size_t)N * C_DIM * sizeof(float), stream);
    hipMemsetAsync(cnt, 0, (size_t)N * sizeof(float), stream);
    aggregate_kernel<<<aggBlocks, 256, 0, stream>>>(hin, edge, edge + E, agg,
                                                    cnt, E);
    gemm_fused<128, 0, true><<<gemmBlocks, 256, ldsLayer, stream>>>(
        agg, cnt, Wl + (size_t)l * C_DIM * C_DIM, hin,
        Wr + (size_t)l * C_DIM * C_DIM, bl + l * C_DIM, bng + l * C_DIM,
        bnb + l * C_DIM, bnm + l * C_DIM, bnv + l * C_DIM, houts[l], N);
    hin = houts[l];
  }

  const int headBlocks = (BATCH + 127) / 128;
  gemm_fused<128, 1, false><<<headBlocks, 256, ldsHead1, stream>>>(
      hin, nullptr, cW1, nullptr, nullptr, cb1, cg1, cbe1, cm1, cv1, h1, BATCH);
  gemm_fused<64, 1, false><<<headBlocks, 256, ldsHead2, stream>>>(
      h1, nullptr, cW2, nullptr, nullptr, cb2, cg2, cbe2, cm2, cv2, h2, BATCH);
  head3_kernel<<<(BATCH + 255) / 256, 256, 0, stream>>>(h2, cW3, cb3, out,
                                                        BATCH);
}